// SPWModulesLayer_52656299049591
// MI455X (gfx1250) — compile-verified
//
#include <hip/hip_runtime.h>

// Problem constants (fixed by the reference)
#define BATCH   64
#define N_IN    800000
#define N_OUTF  2000
#define CAP     400
#define CAP4    (CAP / 4)      // 100 float4 per capsule
#define BN_EPS  1e-5f

typedef float f32x4 __attribute__((ext_vector_type(4)));
typedef float v2f   __attribute__((ext_vector_type(2)));
typedef float v8f   __attribute__((ext_vector_type(8)));

// ---------------------------------------------------------------------------
// Kernel 1: streaming segmented dot-product + bias + relu.
//   grid  = N_OUTF * 8 blocks, 256 threads (8 waves).
//   block (o, btile): 8 waves -> 8 batch rows of the same output column o.
//   Weight segment (400 floats, 1.6KB) staged in LDS once per block.
//   x is nontemporal (streamed once, 204.8 MB); h written regular (L2-hot
//   for kernel 2).
// ---------------------------------------------------------------------------
__global__ __launch_bounds__(256) void capsule_segsum_kernel(
    const float* __restrict__ x, const float* __restrict__ w,
    const float* __restrict__ bias, float* __restrict__ h) {
  __shared__ f32x4 wsm[CAP4];

  const int o     = blockIdx.x >> 3;          // output column 0..1999
  const int bbase = (blockIdx.x & 7) << 3;    // batch tile base (0,8,..,56)
  const int tid   = threadIdx.x;

  if (tid < CAP4)
    wsm[tid] = ((const f32x4*)(w + (size_t)o * CAP))[tid];
  __syncthreads();

  const int wave = tid >> 5;
  const int lane = tid & 31;
  const int b    = bbase + wave;

  const f32x4* xp = (const f32x4*)(x + (size_t)b * N_IN + (size_t)o * CAP);

  // 100 float4 chunks = 3 full wave passes + 4 tail chunks.
  f32x4 a0 = __builtin_nontemporal_load(xp + lane);
  f32x4 a1 = __builtin_nontemporal_load(xp + lane + 32);
  f32x4 a2 = __builtin_nontemporal_load(xp + lane + 64);
  f32x4 acc = a0 * wsm[lane] + a1 * wsm[lane + 32] + a2 * wsm[lane + 64];
  if (lane < 4) {
    f32x4 a3 = __builtin_nontemporal_load(xp + lane + 96);
    acc += a3 * wsm[lane + 96];
  }
  float s = (acc.x + acc.y) + (acc.z + acc.w);

  #pragma unroll
  for (int off = 16; off > 0; off >>= 1)
    s += __shfl_xor(s, off, 32);

  if (lane == 0)
    h[(size_t)b * N_OUTF + o] = fmaxf(s + bias[o], 0.0f);
}

// ---------------------------------------------------------------------------
// Kernel 2: BatchNorm (train-mode, batch axis) + sigmoid gate, in place.
//   grid = N_OUTF/16 = 125 blocks, 256 threads; block owns 16 columns.
//   Batch sums 1^T*H and 1^T*(H.*H) computed with V_WMMA_F32_16X16X4_F32:
//   A = ones(16x4), B = 4x16 tile of h (resp. h^2), 16 K-steps cover 64 rows.
//   Every wave computes the stats redundantly (EXEC all-ones for WMMA),
//   then each wave normalizes its own 8 batch rows after the barrier.
// ---------------------------------------------------------------------------
__global__ __launch_bounds__(256) void bn_gate_kernel(
    float* __restrict__ h, const float* __restrict__ gamma,
    const float* __restrict__ beta, const float* __restrict__ cow) {
  const int n0   = blockIdx.x * 16;
  const int tid  = threadIdx.x;
  const int wave = tid >> 5;
  const int lane = tid & 31;
  const int nl   = lane & 15;
  const int nn   = n0 + nl;            // this lane's column
  const int kb   = (lane >> 4) << 1;   // K rows 0,1 (lanes 0-15) or 2,3 (16-31)

  const v2f ones = {1.0f, 1.0f};
  v8f sum = {0, 0, 0, 0, 0, 0, 0, 0};
  v8f sq  = {0, 0, 0, 0, 0, 0, 0, 0};

  #pragma unroll
  for (int t = 0; t < 16; ++t) {
    const int brow = t * 4 + kb;
    v2f bv;
    bv.x = h[(size_t)brow * N_OUTF + nn];
    bv.y = h[(size_t)(brow + 1) * N_OUTF + nn];
    v2f b2 = bv * bv;
    // D = ones(16x4) x B(4x16) + C  => every D row = column sums of B
    sum = __builtin_amdgcn_wmma_f32_16x16x4_f32(
        false, ones, false, bv, (short)0, sum, false, false);
    sq = __builtin_amdgcn_wmma_f32_16x16x4_f32(
        false, ones, false, b2, (short)0, sq, false, false);
  }

  // Lane L holds colsum for column n0+(L&15) in D[·,L&15] -> element 0.
  const float mean  = sum[0] * (1.0f / BATCH);
  const float var   = sq[0] * (1.0f / BATCH) - mean * mean;
  const float scale = gamma[nn] / sqrtf(var + BN_EPS);
  const float bet   = beta[nn];
  const float sig   = 1.0f / (1.0f + expf(-cow[nn]));

  __syncthreads();  // all stats-phase reads of h done before any write

  // wave w writes rows w*8 .. w*8+7; lane halves split them 4/4.
  const int b0 = wave * 8 + ((lane >> 4) << 2);
  #pragma unroll
  for (int i = 0; i < 4; ++i) {
    const size_t off = (size_t)(b0 + i) * N_OUTF + nn;
    h[off] = ((h[off] - mean) * scale + bet) * sig;
  }
}

// ---------------------------------------------------------------------------
// Launch: inputs = {x, weight, bias, gamma, beta, co_weight, idx}.
// idx is the fixed repeat(arange(2000),400) pattern -> structure hardcoded.
// h lives in d_out (64*2000 f32); kernel 2 rewrites it in place.
// ---------------------------------------------------------------------------
extern "C" void kernel_launch(void* const* d_in, const int* in_sizes, int n_in,
                              void* d_out, int out_size, void* d_ws, size_t ws_size,
                              hipStream_t stream) {
  (void)in_sizes; (void)n_in; (void)out_size; (void)d_ws; (void)ws_size;
  const float* x     = (const float*)d_in[0];
  const float* w     = (const float*)d_in[1];
  const float* bias  = (const float*)d_in[2];
  const float* gamma = (const float*)d_in[3];
  const float* beta  = (const float*)d_in[4];
  const float* cow   = (const float*)d_in[5];
  float* out = (float*)d_out;

  hipLaunchKernelGGL(capsule_segsum_kernel, dim3(N_OUTF * 8), dim3(256), 0,
                     stream, x, w, bias, out);
  hipLaunchKernelGGL(bn_gate_kernel, dim3(N_OUTF / 16), dim3(256), 0,
                     stream, out, gamma, beta, cow);
}